// DEQFusion_403726926348
// MI455X (gfx1250) — compile-verified
//
#include <hip/hip_runtime.h>
#include <hip/hip_bf16.h>
#include <math.h>

#define C_DIM 128
#define D_TOT 384
#define LDX   132          // padded LDS row stride (floats) to avoid bank conflicts
#define GN_EPS 1e-5f
#define AND_LAM 1e-4f

typedef float v2f __attribute__((ext_vector_type(2)));
typedef float v8f __attribute__((ext_vector_type(8)));

// ---------------------------------------------------------------------------
// 16x16 output tile of  A(16x128, LDS) @ W^T  with W (128x128 row-major, global)
// via V_WMMA_F32_16X16X4_F32, K accumulated 4 at a time (32 wmma ops).
// A frag (16x4 f32): lane l holds M=l&15, K = k + 2*(l>>4) .. +1
// B frag (4x16 f32): lane l holds N=nbase+(l&15), same K pair  (B[k][n] = W[n][k])
// ---------------------------------------------------------------------------
__device__ __forceinline__ v8f mm_tile_xwT(const float* __restrict__ xt,
                                           const float* __restrict__ w,
                                           int nbase, int lane) {
  v8f c = {};
  const int m    = lane & 15;
  const int koff = (lane >> 4) << 1;
  const float* wr = w + (size_t)(nbase + m) * C_DIM;
  const float* xr = xt + m * LDX;
#pragma unroll 8
  for (int k = 0; k < C_DIM; k += 4) {
    const int kk = k + koff;
    v2f a; a[0] = xr[kk]; a[1] = xr[kk + 1];
    v2f b; b[0] = wr[kk]; b[1] = wr[kk + 1];
    c = __builtin_amdgcn_wmma_f32_16x16x4_f32(false, a, false, b, (short)0, c,
                                              false, false);
  }
  return c;
}

// Per-(row,group) mean / rsqrt(var+eps); groups of 32 channels, 16 rows.
__device__ __forceinline__ void gn_stats(const float* __restrict__ t,
                                         float* __restrict__ mu,
                                         float* __restrict__ rs) {
  const int tid = threadIdx.x;
  if (tid < 64) {
    const int row = tid >> 2, g = tid & 3;
    const float* p = t + row * LDX + (g << 5);
    float s = 0.f, s2 = 0.f;
#pragma unroll
    for (int j = 0; j < 32; ++j) { float v = p[j]; s += v; s2 += v * v; }
    const float m  = s * 0.03125f;
    const float var = s2 * 0.03125f - m * m;
    mu[tid] = m;
    rs[tid] = rsqrtf(var + GN_EPS);
  }
}

struct SB { const float *w1,*b1,*g1w,*g1b,*w2,*b2,*g2w,*g2b,*g3w,*g3b,*inj; };
struct FB { const float *gw,*gb,*fw,*fbv,*g3w,*g3b; };

// ---------------------------------------------------------------------------
// simple_block: out[:, coff:coff+128] = gn3(relu(gn2(u@w2^T+b2+inj) + x)),
//               u = relu(gn1(x@w1^T+b1)),  x = z[:, coff:coff+128]
// grid.x = B/16 row tiles, grid.y = {0,1} selects block params / column slab
// ---------------------------------------------------------------------------
__global__ __launch_bounds__(256) void simple_block_kernel(SB P0, SB P1,
    const float* __restrict__ z, float* __restrict__ out, int B) {
  __shared__ float xs[16 * LDX];
  __shared__ float ts[16 * LDX];
  __shared__ float us[16 * LDX];
  __shared__ float mu[64], rs[64];

  const SB  P    = (blockIdx.y == 0) ? P0 : P1;
  const int coff = blockIdx.y * C_DIM;
  const int row0 = blockIdx.x * 16;
  const int tid  = threadIdx.x;
  const int lane = tid & 31;
  const int nbase = (tid >> 5) << 4;

  for (int i = tid; i < 16 * C_DIM; i += 256) {
    int r = i >> 7, c = i & 127;
    xs[r * LDX + c] = z[(size_t)(row0 + r) * D_TOT + coff + c];
  }
  __syncthreads();

  { // t = x @ w1^T + b1
    v8f c = mm_tile_xwT(xs, P.w1, nbase, lane);
    const int col = nbase + (lane & 15);
    const int m0  = (lane >> 4) << 3;
    const float bb = P.b1[col];
#pragma unroll
    for (int r = 0; r < 8; ++r) ts[(m0 + r) * LDX + col] = c[r] + bb;
  }
  __syncthreads();
  gn_stats(ts, mu, rs);
  __syncthreads();
  for (int i = tid; i < 16 * C_DIM; i += 256) {           // gn1 + relu
    int r = i >> 7, c = i & 127, g = c >> 5;
    float v = (ts[r * LDX + c] - mu[(r << 2) + g]) * rs[(r << 2) + g];
    ts[r * LDX + c] = fmaxf(v * P.g1w[c] + P.g1b[c], 0.f);
  }
  __syncthreads();

  { // v = u @ w2^T + b2 + inj
    v8f c = mm_tile_xwT(ts, P.w2, nbase, lane);
    const int col = nbase + (lane & 15);
    const int m0  = (lane >> 4) << 3;
    const float bb = P.b2[col];
#pragma unroll
    for (int r = 0; r < 8; ++r)
      us[(m0 + r) * LDX + col] =
          c[r] + bb + P.inj[(size_t)(row0 + m0 + r) * C_DIM + col];
  }
  __syncthreads();
  gn_stats(us, mu, rs);
  __syncthreads();
  for (int i = tid; i < 16 * C_DIM; i += 256) {           // gn2, +x, relu
    int r = i >> 7, c = i & 127, g = c >> 5;
    float v = (us[r * LDX + c] - mu[(r << 2) + g]) * rs[(r << 2) + g];
    v = v * P.g2w[c] + P.g2b[c] + xs[r * LDX + c];
    us[r * LDX + c] = fmaxf(v, 0.f);
  }
  __syncthreads();
  gn_stats(us, mu, rs);
  __syncthreads();
  for (int i = tid; i < 16 * C_DIM; i += 256) {           // gn3 -> out
    int r = i >> 7, c = i & 127, g = c >> 5;
    float v = (us[r * LDX + c] - mu[(r << 2) + g]) * rs[(r << 2) + g];
    out[(size_t)(row0 + r) * D_TOT + coff + c] = v * P.g3w[c] + P.g3b[c];
  }
}

// ---------------------------------------------------------------------------
// fusion_block: s = sum_p x*((o_p + x)@gw^T + gb); out = relu(gn3(s@fw^T+fb+fus))
// reads o0,o1 from out buffer cols [0,256), writes cols [256,384)
// ---------------------------------------------------------------------------
__global__ __launch_bounds__(256) void fusion_block_kernel(FB P,
    const float* __restrict__ z, const float* __restrict__ o01,
    const float* __restrict__ fus, float* __restrict__ out, int B) {
  __shared__ float xs[16 * LDX];
  __shared__ float ts[16 * LDX];
  __shared__ float us[16 * LDX];
  __shared__ float mu[64], rs[64];

  const int row0 = blockIdx.x * 16;
  const int tid  = threadIdx.x;
  const int lane = tid & 31;
  const int nbase = (tid >> 5) << 4;

  for (int i = tid; i < 16 * C_DIM; i += 256) {
    int r = i >> 7, c = i & 127;
    xs[r * LDX + c] = z[(size_t)(row0 + r) * D_TOT + 2 * C_DIM + c];
  }
  __syncthreads();

  for (int pass = 0; pass < 2; ++pass) {
    for (int i = tid; i < 16 * C_DIM; i += 256) {
      int r = i >> 7, c = i & 127;
      ts[r * LDX + c] = xs[r * LDX + c] +
                        o01[(size_t)(row0 + r) * D_TOT + pass * C_DIM + c];
    }
    __syncthreads();
    v8f c = mm_tile_xwT(ts, P.gw, nbase, lane);
    const int col = nbase + (lane & 15);
    const int m0  = (lane >> 4) << 3;
    const float bb = P.gb[col];
#pragma unroll
    for (int r = 0; r < 8; ++r) {
      float sv = xs[(m0 + r) * LDX + col] * (c[r] + bb);
      if (pass == 0) us[(m0 + r) * LDX + col] = sv;
      else           us[(m0 + r) * LDX + col] += sv;
    }
    __syncthreads();
  }

  { // out = s @ fw^T + fb + fus
    v8f c = mm_tile_xwT(us, P.fw, nbase, lane);
    const int col = nbase + (lane & 15);
    const int m0  = (lane >> 4) << 3;
    const float bb = P.fbv[col];
#pragma unroll
    for (int r = 0; r < 8; ++r)
      ts[(m0 + r) * LDX + col] =
          c[r] + bb + fus[(size_t)(row0 + m0 + r) * C_DIM + col];
  }
  __syncthreads();
  gn_stats(ts, mu, rs);
  __syncthreads();
  for (int i = tid; i < 16 * C_DIM; i += 256) {           // relu(gn3(.))
    int r = i >> 7, c = i & 127, g = c >> 5;
    float v = (ts[r * LDX + c] - mu[(r << 2) + g]) * rs[(r << 2) + g];
    v = v * P.g3w[c] + P.g3b[c];
    out[(size_t)(row0 + r) * D_TOT + 2 * C_DIM + c] = fmaxf(v, 0.f);
  }
}

// ---------------------------------------------------------------------------
// Anderson Gram: Hb[b] = [[0,1..],[1, G G^T + lam I]], wave32 per batch row.
// ---------------------------------------------------------------------------
__global__ __launch_bounds__(128) void gram_kernel(const float* __restrict__ Xb,
    const float* __restrict__ Fb, float* __restrict__ Hb, int n, int B) {
  const int wv   = (int)((blockIdx.x * (size_t)blockDim.x + threadIdx.x) >> 5);
  const int lane = threadIdx.x & 31;
  if (wv >= B) return;
  float G[6][12];
#pragma unroll
  for (int i = 0; i < 6; ++i) {
    if (i < n) {
      const float* fp = Fb + ((size_t)i * B + wv) * D_TOT;
      const float* xp = Xb + ((size_t)i * B + wv) * D_TOT;
#pragma unroll
      for (int j = 0; j < 12; ++j) {
        int d = lane + (j << 5);
        G[i][j] = fp[d] - xp[d];
      }
    } else {
#pragma unroll
      for (int j = 0; j < 12; ++j) G[i][j] = 0.f;
    }
  }
  float* hb = Hb + (size_t)wv * 49;
  const int np1 = n + 1;
#pragma unroll
  for (int i = 0; i < 6; ++i) {
#pragma unroll
    for (int j = i; j < 6; ++j) {
      if (j < n) {
        float s = 0.f;
#pragma unroll
        for (int t = 0; t < 12; ++t) s += G[i][t] * G[j][t];
#pragma unroll
        for (int off = 16; off > 0; off >>= 1) s += __shfl_xor(s, off, 32);
        if (lane == 0) {
          float v = s + ((i == j) ? AND_LAM : 0.f);
          hb[(i + 1) * np1 + (j + 1)] = v;
          if (i != j) hb[(j + 1) * np1 + (i + 1)] = v;
        }
      }
    }
  }
  if (lane == 0) {
    hb[0] = 0.f;
    for (int i = 1; i <= n; ++i) { hb[i] = 1.f; hb[i * np1] = 1.f; }
  }
}

// (n+1)x(n+1) solve, one thread per batch row; alpha = sol[1..n]
__global__ __launch_bounds__(128) void solve_kernel(const float* __restrict__ Hb,
    float* __restrict__ alpha, int n, int B) {
  const int b = blockIdx.x * blockDim.x + threadIdx.x;
  if (b >= B) return;
  const int np1 = n + 1;
  float A[7][8];
  const float* hb = Hb + (size_t)b * 49;
  for (int i = 0; i < np1; ++i) {
    for (int j = 0; j < np1; ++j) A[i][j] = hb[i * np1 + j];
    A[i][np1] = (i == 0) ? 1.f : 0.f;
  }
  for (int c = 0; c < np1; ++c) {
    int piv = c; float best = fabsf(A[c][c]);
    for (int r = c + 1; r < np1; ++r) {
      float v = fabsf(A[r][c]);
      if (v > best) { best = v; piv = r; }
    }
    if (piv != c)
      for (int j = c; j <= np1; ++j) { float t = A[c][j]; A[c][j] = A[piv][j]; A[piv][j] = t; }
    float inv = 1.f / A[c][c];
    for (int j = c; j <= np1; ++j) A[c][j] *= inv;
    for (int r = 0; r < np1; ++r) if (r != c) {
      float f = A[r][c];
      for (int j = c; j <= np1; ++j) A[r][j] -= f * A[c][j];
    }
  }
  for (int i = 0; i < n; ++i) alpha[(size_t)b * 6 + i] = A[i + 1][np1];
}

// xk = sum_i alpha_i * F_i   (beta == 1.0); also X[slot] = xk, z = xk
__global__ void combine_kernel(const float* __restrict__ Fb,
    const float* __restrict__ alpha, float* __restrict__ z,
    float* __restrict__ Xslot, int n, size_t total) {
  size_t idx = (size_t)blockIdx.x * blockDim.x + threadIdx.x;
  if (idx >= total) return;
  size_t b = idx / D_TOT;
  float s = 0.f;
#pragma unroll
  for (int i = 0; i < 6; ++i)
    if (i < n) s += alpha[b * 6 + i] * Fb[(size_t)i * total + idx];
  z[idx] = s;
  Xslot[idx] = s;
}

__global__ void fill_kernel(float* p, float v, size_t n) {
  size_t i = (size_t)blockIdx.x * blockDim.x + threadIdx.x;
  if (i < n) p[i] = v;
}
__global__ void copy2_kernel(const float* __restrict__ s, float* d1, float* d2, size_t n) {
  size_t i = (size_t)blockIdx.x * blockDim.x + threadIdx.x;
  if (i < n) { float v = s[i]; d1[i] = v; d2[i] = v; }
}
__global__ void fus_kernel(const float* __restrict__ f0, const float* __restrict__ f1,
                           const float* mw0, const float* mw1, const float* bias,
                           float* __restrict__ fus, size_t n) {
  size_t i = (size_t)blockIdx.x * blockDim.x + threadIdx.x;
  if (i < n) fus[i] = mw0[0] * f0[i] + mw1[0] * f1[i] + bias[0];
}
__global__ void extract_kernel(const float* __restrict__ z, float* __restrict__ out, int B) {
  int i = blockIdx.x * blockDim.x + threadIdx.x;
  if (i < B * C_DIM) {
    int b = i >> 7, c = i & 127;
    out[i] = z[(size_t)b * D_TOT + 2 * C_DIM + c];
  }
}

// ---------------------------------------------------------------------------
extern "C" void kernel_launch(void* const* d_in, const int* in_sizes, int n_in,
                              void* d_out, int out_size, void* d_ws, size_t ws_size,
                              hipStream_t stream) {
  const float* f0 = (const float*)d_in[0];
  const float* f1 = (const float*)d_in[1];
  // params pytree (sorted dict keys):
  // b0 -> 2:b1 3:b2 4:g1b 5:g1w 6:g2b 7:g2w 8:g3b 9:g3w 10:w1 11:w2
  // b1 -> 12..21 (same order) ; bias -> 22
  // fb -> 23:fb 24:fw 25:g3b 26:g3w 27:gb 28:gw ; mw0 -> 29 ; mw1 -> 30
  auto F = [&](int i) { return (const float*)d_in[i]; };
  SB P0 { F(10), F(2),  F(5),  F(4),  F(11), F(3),  F(7),  F(6),  F(9),  F(8),  f0 };
  SB P1 { F(20), F(12), F(15), F(14), F(21), F(13), F(17), F(16), F(19), F(18), f1 };
  FB PF { F(28), F(27), F(24), F(23), F(26), F(25) };
  const float* bias = F(22);
  const float* mw0  = F(29);
  const float* mw1  = F(30);

  const int    B     = in_sizes[0] / C_DIM;
  const size_t total = (size_t)B * D_TOT;

  float* ws   = (float*)d_ws;
  float* fusb = ws;                         // B*128
  float* z    = fusb + (size_t)B * C_DIM;   // B*384
  float* Xb   = z + total;                  // 6*B*384
  float* Fb   = Xb + 6 * total;             // 6*B*384
  float* Hb   = Fb + 6 * total;             // B*49
  float* al   = Hb + (size_t)B * 49;        // B*6

  auto feval = [&](float* dst) {
    dim3 g(B / 16, 2);
    simple_block_kernel<<<g, 256, 0, stream>>>(P0, P1, z, dst, B);
    fusion_block_kernel<<<B / 16, 256, 0, stream>>>(PF, z, dst, fusb, dst, B);
  };

  { // fus = mw0*f0 + mw1*f1 + bias
    size_t n = (size_t)B * C_DIM;
    fus_kernel<<<(int)((n + 255) / 256), 256, 0, stream>>>(f0, f1, mw0, mw1, bias, fusb, n);
  }
  const int gt = (int)((total + 255) / 256);
  fill_kernel<<<gt, 256, 0, stream>>>(z, 0.f, total);     // x0 = 0
  fill_kernel<<<gt, 256, 0, stream>>>(Xb, 0.f, total);    // X[0] = 0
  feval(Fb);                                              // F[0] = f(0)
  copy2_kernel<<<gt, 256, 0, stream>>>(Fb, Xb + total, z, total); // X[1]=F[0], z=F[0]
  feval(Fb + total);                                      // F[1] = f(z)

  for (int k = 2; k < 55; ++k) {
    const int n    = k < 6 ? k : 6;
    const int slot = k % 6;
    gram_kernel<<<(B * 32 + 127) / 128, 128, 0, stream>>>(Xb, Fb, Hb, n, B);
    solve_kernel<<<(B + 127) / 128, 128, 0, stream>>>(Hb, al, n, B);
    combine_kernel<<<gt, 256, 0, stream>>>(Fb, al, z, Xb + (size_t)slot * total, n, total);
    if (k < 54) feval(Fb + (size_t)slot * total);         // F[k%6] = f(xk); last one unused
  }
  extract_kernel<<<(B * C_DIM + 255) / 256, 256, 0, stream>>>(z, (float*)d_out, B);
}